// GNNNetwork_42434276884572
// MI455X (gfx1250) — compile-verified
//
#include <hip/hip_runtime.h>
#include <hip/hip_bf16.h>
#include <math.h>

typedef float v2f __attribute__((ext_vector_type(2)));
typedef float v8f __attribute__((ext_vector_type(8)));
typedef int   v4i __attribute__((ext_vector_type(4)));

#if defined(__gfx1250__) && __has_builtin(__builtin_amdgcn_global_load_async_to_lds_b128)
#define USE_ASYNC_LDS 1
#else
#define USE_ASYNC_LDS 0
#endif

__device__ __forceinline__ void wait_asynccnt0() {
#if __has_builtin(__builtin_amdgcn_s_wait_asynccnt)
    __builtin_amdgcn_s_wait_asynccnt(0);
#else
    asm volatile("s_wait_asynccnt 0x0" ::: "memory");
#endif
}

// ---------------------------------------------------------------------------
// GEMM: C[nrows x 128] = A[nrows x 128] @ W[128 x 128], fp32, WMMA 16x16x4.
// Block = 256 threads = 8 waves. W staged in LDS (64 KB) via async-to-LDS
// (ASYNCcnt path) when available. Wave w computes the 16-row stripe
// [blockIdx*128 + 16w, +16) across all 8 column tiles.
// A-frag layout (32-bit A 16x4): lanes 0-15 -> (m=lane, k0,k0+1),
//                                lanes 16-31 -> (m=lane-16, k0+2,k0+3).
// B-frag mirrors it (row of B striped across lanes; lane = n).
// C/D layout: VGPR r -> M=r (lanes 0-15), M=r+8 (lanes 16-31); N = lane&15.
// ---------------------------------------------------------------------------
__global__ __launch_bounds__(256) void gemm_x128(const float* __restrict__ A,
                                                 const float* __restrict__ W,
                                                 float* __restrict__ C,
                                                 int nrows)
{
    __shared__ float sW[128 * 128];

#if USE_ASYNC_LDS
    {
        // 4096 x b128 chunks; each thread issues 16 async LDS loads, then
        // drains its ASYNCcnt before the block barrier.
        typedef __attribute__((address_space(1))) v4i gv4i_t;
        typedef __attribute__((address_space(3))) v4i lv4i_t;
        gv4i_t* gp = (gv4i_t*)W;
        lv4i_t* lp = (lv4i_t*)sW;
        for (int i = threadIdx.x; i < (128 * 128) / 4; i += 256)
            __builtin_amdgcn_global_load_async_to_lds_b128(gp + i, lp + i, 0, 0);
        wait_asynccnt0();
    }
#else
    for (int i = threadIdx.x; i < (128 * 128) / 4; i += 256)
        ((float4*)sW)[i] = ((const float4*)W)[i];
#endif
    __syncthreads();

    const int wave = threadIdx.x >> 5;
    const int lane = threadIdx.x & 31;
    const int row0 = blockIdx.x * 128 + wave * 16;
    if (row0 >= nrows) return;              // wave-uniform: EXEC stays all-1s

    const int m    = lane & 15;
    const int koff = (lane >> 4) << 1;      // 0 or 2
    const int ncol = lane & 15;
    const float* arow = A + (size_t)(row0 + m) * 128;

    v8f acc[8] = {};
    for (int k0 = 0; k0 < 128; k0 += 4) {
        v2f a = *(const v2f*)(arow + k0 + koff);     // 8B aligned global load
        const float* wp = &sW[(k0 + koff) * 128 + ncol];
#pragma unroll
        for (int t = 0; t < 8; ++t) {
            v2f b;
            b.x = wp[t * 16];
            b.y = wp[128 + t * 16];
            acc[t] = __builtin_amdgcn_wmma_f32_16x16x4_f32(
                false, a, false, b, (short)0, acc[t], false, false);
        }
    }

#pragma unroll
    for (int t = 0; t < 8; ++t) {
#pragma unroll
        for (int r = 0; r < 8; ++r) {
            const int mm = r + ((lane >> 4) << 3);
            C[(size_t)(row0 + mm) * 128 + t * 16 + ncol] = acc[t][r];
        }
    }
}

// ---------------------------------------------------------------------------
// Small utility kernels
// ---------------------------------------------------------------------------
__global__ void fill_f32(float* p, float v, int n) {
    int i = blockIdx.x * blockDim.x + threadIdx.x;
    if (i < n) p[i] = v;
}

__global__ void deg_count(const int* __restrict__ dst, float* __restrict__ deg, int nE) {
    int e = blockIdx.x * blockDim.x + threadIdx.x;
    if (e < nE) atomicAdd(&deg[dst[e]], 1.0f);
}

__global__ void deg_to_dinv(float* deg, int n) {
    int i = blockIdx.x * blockDim.x + threadIdx.x;
    if (i < n) deg[i] = rsqrtf(deg[i]);     // deg >= 1 always (self loops)
}

// out[i][c] = xw[i][c] * dinv[i]^2   (self-loop message)
__global__ void self_loop_init(const float* __restrict__ xw,
                               const float* __restrict__ dinv,
                               float* __restrict__ out, int total) {
    int i = blockIdx.x * blockDim.x + threadIdx.x;
    if (i < total) {
        float s = dinv[i >> 7];
        out[i] = xw[i] * s * s;
    }
}

// One wave per edge: coalesced float4 gather of xw[src], scale by
// dinv[src]*dinv[dst], 4 fp32 atomics per lane into out[dst].
// xw (51 MB) and out (51 MB) both fit in the 192 MB L2.
__global__ __launch_bounds__(256) void edge_scatter(const int* __restrict__ src,
                                                    const int* __restrict__ dst,
                                                    const float* __restrict__ dinv,
                                                    const float* __restrict__ xw,
                                                    float* __restrict__ out, int nE)
{
    int e = blockIdx.x * 8 + (threadIdx.x >> 5);
    if (e >= nE) return;
    int lane = threadIdx.x & 31;
    int s = src[e], d = dst[e];
    float nrm = dinv[s] * dinv[d];
    float4 v = *(const float4*)(xw + (size_t)s * 128 + lane * 4);
    float* o = out + (size_t)d * 128 + lane * 4;
    atomicAdd(o + 0, v.x * nrm);
    atomicAdd(o + 1, v.y * nrm);
    atomicAdd(o + 2, v.z * nrm);
    atomicAdd(o + 3, v.w * nrm);
}

// Per-channel sum & sumsq partials -> atomics into sums[0:128], sums[128:256]
__global__ __launch_bounds__(256) void bn_stats(const float* __restrict__ x,
                                                float* __restrict__ sums, int n)
{
    __shared__ float ssum[256], ssq[256];
    int c = threadIdx.x & 127;
    int half = threadIdx.x >> 7;
    float s = 0.f, q = 0.f;
    for (int r = blockIdx.x * 2 + half; r < n; r += gridDim.x * 2) {
        float v = x[(size_t)r * 128 + c];
        s += v; q += v * v;
    }
    ssum[threadIdx.x] = s; ssq[threadIdx.x] = q;
    __syncthreads();
    if (half == 0) {
        atomicAdd(&sums[c],       ssum[c] + ssum[c + 128]);
        atomicAdd(&sums[128 + c], ssq[c]  + ssq[c + 128]);
    }
}

// scsh[c] = gamma*rstd ; scsh[128+c] = beta - mean*gamma*rstd
__global__ void bn_finalize(const float* __restrict__ sums,
                            const float* __restrict__ gamma,
                            const float* __restrict__ beta,
                            float* __restrict__ scsh, float n)
{
    int c = threadIdx.x;
    if (c >= 128) return;
    float mean  = sums[c] / n;
    float var   = sums[128 + c] / n - mean * mean;
    float rstd  = rsqrtf(var + 1e-5f);
    float scale = gamma[c] * rstd;
    scsh[c]       = scale;
    scsh[128 + c] = beta[c] - mean * scale;
}

// x = relu(x*scale + shift), in place
__global__ void bn_relu_apply(float* x, const float* __restrict__ scsh, int total) {
    int i = blockIdx.x * blockDim.x + threadIdx.x;
    if (i < total) {
        int c = i & 127;
        float v = x[i] * scsh[c] + scsh[128 + c];
        x[i] = fmaxf(v, 0.f);
    }
}

// blp[o] = bl[o] + sum_c b2[c] * Wl[128+c][o]   (fold conv2 bias into linear)
__global__ void bias_fold(const float* __restrict__ b2, const float* __restrict__ Wl,
                          const float* __restrict__ bl, float* __restrict__ blp)
{
    int o = threadIdx.x;
    if (o >= 40) return;
    float s = bl[o];
    for (int c = 0; c < 128; ++c) s += b2[c] * Wl[(128 + c) * 40 + o];
    blp[o] = s;
}

// Fused concat -> linear(256x40) -> log_softmax, one node per thread,
// Wl (40 KB) broadcast from LDS.
__global__ __launch_bounds__(256) void final_head(const float* __restrict__ x1,
                                                  const float* __restrict__ agg2,
                                                  const float* __restrict__ Wl,
                                                  const float* __restrict__ blp,
                                                  float* __restrict__ out, int n)
{
    __shared__ float sW[256 * 40];
    __shared__ float sb[40];
    for (int i = threadIdx.x; i < 256 * 40; i += 256) sW[i] = Wl[i];
    if (threadIdx.x < 40) sb[threadIdx.x] = blp[threadIdx.x];
    __syncthreads();

    int node = blockIdx.x * 256 + threadIdx.x;
    if (node >= n) return;

    float acc[40];
#pragma unroll
    for (int o = 0; o < 40; ++o) acc[o] = sb[o];

    const float4* xa = (const float4*)(x1   + (size_t)node * 128);
    const float4* xb = (const float4*)(agg2 + (size_t)node * 128);
    for (int c4 = 0; c4 < 32; ++c4) {
        float4 v = xa[c4];
#pragma unroll
        for (int j = 0; j < 4; ++j) {
            float xv = (&v.x)[j];
            const float* w = &sW[(c4 * 4 + j) * 40];
#pragma unroll
            for (int o = 0; o < 40; ++o) acc[o] += xv * w[o];
        }
    }
    for (int c4 = 0; c4 < 32; ++c4) {
        float4 v = xb[c4];
#pragma unroll
        for (int j = 0; j < 4; ++j) {
            float xv = (&v.x)[j];
            const float* w = &sW[(128 + c4 * 4 + j) * 40];
#pragma unroll
            for (int o = 0; o < 40; ++o) acc[o] += xv * w[o];
        }
    }

    float mx = acc[0];
#pragma unroll
    for (int o = 1; o < 40; ++o) mx = fmaxf(mx, acc[o]);
    float se = 0.f;
#pragma unroll
    for (int o = 0; o < 40; ++o) se += expf(acc[o] - mx);
    float lse = logf(se) + mx;

    float* op = out + (size_t)node * 40;
#pragma unroll
    for (int o = 0; o < 40; ++o) op[o] = acc[o] - lse;
}

// ---------------------------------------------------------------------------
static inline int cdiv(int a, int b) { return (a + b - 1) / b; }

extern "C" void kernel_launch(void* const* d_in, const int* in_sizes, int n_in,
                              void* d_out, int out_size, void* d_ws, size_t ws_size,
                              hipStream_t stream)
{
    const float* x0    = (const float*)d_in[0];
    const int*   eidx  = (const int*)d_in[1];
    const float* W1    = (const float*)d_in[2];
    // d_in[3] = b1: cancels exactly inside BatchNorm (pure mean shift)
    const float* gamma = (const float*)d_in[4];
    const float* beta  = (const float*)d_in[5];
    const float* W2    = (const float*)d_in[6];
    const float* b2    = (const float*)d_in[7];
    const float* Wl    = (const float*)d_in[8];
    const float* bl    = (const float*)d_in[9];

    const int N = in_sizes[0] / 128;   // 100000
    const int E = in_sizes[1] / 2;     // 1600000
    const int* src = eidx;
    const int* dst = eidx + E;

    // Workspace layout (fp32): xw | x1 | agg2 | dinv | sums | scsh | blp
    float* xw   = (float*)d_ws;
    float* x1   = xw   + (size_t)N * 128;
    float* agg2 = x1   + (size_t)N * 128;
    float* dinv = agg2 + (size_t)N * 128;
    float* sums = dinv + N;
    float* scsh = sums + 256;
    float* blp  = scsh + 256;

    const int tot = N * 128;
    const int gemmBlocks = cdiv(N, 128);

    // --- degrees / symmetric norm ---
    fill_f32<<<cdiv(N, 256), 256, 0, stream>>>(dinv, 1.0f, N);
    deg_count<<<cdiv(E, 256), 256, 0, stream>>>(dst, dinv, E);
    deg_to_dinv<<<cdiv(N, 256), 256, 0, stream>>>(dinv, N);

    // --- conv1: xw = x0 @ W1 ; aggregate into x1-buffer ---
    gemm_x128<<<gemmBlocks, 256, 0, stream>>>(x0, W1, xw, N);
    self_loop_init<<<cdiv(tot, 256), 256, 0, stream>>>(xw, dinv, x1, tot);
    edge_scatter<<<cdiv(E, 8), 256, 0, stream>>>(src, dst, dinv, xw, x1, E);

    // --- BatchNorm (training stats) + ReLU, in place -> x1 ---
    fill_f32<<<1, 256, 0, stream>>>(sums, 0.0f, 256);
    bn_stats<<<512, 256, 0, stream>>>(x1, sums, N);
    bn_finalize<<<1, 128, 0, stream>>>(sums, gamma, beta, scsh, (float)N);
    bn_relu_apply<<<cdiv(tot, 256), 256, 0, stream>>>(x1, scsh, tot);

    // --- conv2: xw = x1 @ W2 ; aggregate into agg2 (b2 folded into head) ---
    gemm_x128<<<gemmBlocks, 256, 0, stream>>>(x1, W2, xw, N);
    self_loop_init<<<cdiv(tot, 256), 256, 0, stream>>>(xw, dinv, agg2, tot);
    edge_scatter<<<cdiv(E, 8), 256, 0, stream>>>(src, dst, dinv, xw, agg2, E);

    // --- head: concat -> linear -> log_softmax ---
    bias_fold<<<1, 64, 0, stream>>>(b2, Wl, bl, blp);
    final_head<<<cdiv(N, 256), 256, 0, stream>>>(x1, agg2, Wl, blp,
                                                 (float*)d_out, N);
}